// PointerRubricModel_44246753084249
// MI455X (gfx1250) — compile-verified
//
#include <hip/hip_runtime.h>
#include <hip/hip_bf16.h>
#include <math.h>

#define Bdim 32
#define Sdim 2048
#define Hdim 1024
#define Rdim 64

typedef __attribute__((ext_vector_type(2))) float v2f;
typedef __attribute__((ext_vector_type(8))) float v8f;

// ---------------------------------------------------------------------------
// Kernel 1: answer-span mean pooling.  one block per batch row.
// answer_emb[b,h] = mean_{s in [s0,s1)} seq[b,s,h]
// ---------------------------------------------------------------------------
__global__ void answer_pool_kernel(const float* __restrict__ seq,
                                   const int* __restrict__ answer_span,
                                   float* __restrict__ answer_emb) {
    const int b  = blockIdx.x;
    const int s0 = answer_span[2 * b];
    const int s1 = answer_span[2 * b + 1];
    int len = s1 - s0;
    if (len < 1) len = 1;
    const float inv = 1.0f / (float)len;
    for (int h = threadIdx.x; h < Hdim; h += blockDim.x) {
        float acc = 0.0f;
        const float* base = seq + ((size_t)b * Sdim) * Hdim + h;
        for (int s = s0; s < s1; ++s) {
            acc += base[(size_t)s * Hdim];
        }
        answer_emb[b * Hdim + h] = acc * inv;
    }
}

// ---------------------------------------------------------------------------
// Kernel 2: u[b,h] = sum_k W[h,k] * answer_emb[b,k]
// i.e. U[32,1024] = A[32,1024] @ W^T[1024,1024], done with
// V_WMMA_F32_16X16X4_F32 (exact f32 — matches the f32 reference).
// One wave (32 threads) per 16x16 output tile; K-loop of 256 WMMAs.
//
// A 16x4 f32 layout : lane L holds M = L&15 ; VGPR0 = K(base+ (L<16?0:2)),
//                     VGPR1 = next K  -> contiguous float2 per lane.
// B 4x16 f32 layout : lane L holds N = L&15 ; same K striping; element
//                     B[k][n] = W[n][k] -> contiguous float2 from W row n.
// D 16x16 f32 layout: VGPR j -> row mBase + j + (L<16?0:8), col nBase+(L&15).
// ---------------------------------------------------------------------------
__global__ void bilinear_u_wmma_kernel(const float* __restrict__ answer_emb, // [32,1024]
                                       const float* __restrict__ W,          // [1024,1024]
                                       float* __restrict__ U) {              // [32,1024]
    const int lane  = threadIdx.x;           // 0..31, one wave per block
    const int nBase = blockIdx.x * 16;       // over H (output cols)
    const int mBase = blockIdx.y * 16;       // over B (output rows)

    const int mRow = mBase + (lane & 15);
    const int nRow = nBase + (lane & 15);
    const int kOff = (lane < 16) ? 0 : 2;

    const float* aRow = answer_emb + (size_t)mRow * Hdim + kOff;
    const float* wRow = W          + (size_t)nRow * Hdim + kOff;

    v8f acc = {};
    for (int k = 0; k < Hdim; k += 4) {
        v2f av = *(const v2f*)(aRow + k);
        v2f bv = *(const v2f*)(wRow + k);
        acc = __builtin_amdgcn_wmma_f32_16x16x4_f32(
            /*neg_a=*/false, av, /*neg_b=*/false, bv,
            /*c_mod=*/(short)0, acc, /*reuse_a=*/false, /*reuse_b=*/false);
    }

    const int mAdd = (lane < 16) ? 0 : 8;
    const int nCol = nBase + (lane & 15);
#pragma unroll
    for (int j = 0; j < 8; ++j) {
        U[(size_t)(mBase + mAdd + j) * Hdim + nCol] = acc[j];
    }
}

// ---------------------------------------------------------------------------
// Kernel 3: rubric scores + masked softmax.  One 1024-thread block per batch.
// score[b,r] = (1/len_r) * sum_{s in span_r} dot(seq[b,s,:], u[b,:]) + bias
// masked -> -inf ; softmax over R=64.
// u[b,:] is staged in LDS; one wave handles two rubrics.
// ---------------------------------------------------------------------------
__global__ void scores_softmax_kernel(const float* __restrict__ seq,
                                      const float* __restrict__ U,
                                      const int* __restrict__ rubric_span,   // [B,R,2]
                                      const unsigned char* __restrict__ rubric_mask, // [B,R] bool
                                      const float* __restrict__ bias,        // [1]
                                      float* __restrict__ out) {             // [B,R]
    __shared__ float u_s[Hdim];
    __shared__ float sc_s[Rdim];

    const int b   = blockIdx.x;
    const int tid = threadIdx.x;           // 0..1023
    const int wave = tid >> 5;
    const int lane = tid & 31;

    u_s[tid] = U[(size_t)b * Hdim + tid];
    __syncthreads();

    const float biasv = bias[0];

    // 32 waves, 64 rubrics -> each wave does r = wave and r = wave+32
    for (int r = wave; r < Rdim; r += 32) {
        const int idx = (b * Rdim + r);
        const int s0 = rubric_span[2 * idx];
        const int s1 = rubric_span[2 * idx + 1];
        const bool valid = (rubric_mask[idx] != 0);

        float acc = 0.0f;
        for (int s = s0; s < s1; ++s) {
            const float* row = seq + ((size_t)b * Sdim + s) * Hdim;
#pragma unroll
            for (int h0 = 0; h0 < Hdim; h0 += 32) {
                const int h = h0 + lane;
                acc = fmaf(row[h], u_s[h], acc);
            }
        }
        // wave32 butterfly reduction
#pragma unroll
        for (int off = 16; off > 0; off >>= 1) {
            acc += __shfl_xor(acc, off, 32);
        }
        if (lane == 0) {
            int len = s1 - s0;
            if (len < 1) len = 1;
            sc_s[r] = valid ? (acc / (float)len + biasv) : -INFINITY;
        }
    }
    __syncthreads();

    // masked softmax over R=64 (redundant per-thread, trivially cheap)
    if (tid < Rdim) {
        float m = -INFINITY;
#pragma unroll
        for (int i = 0; i < Rdim; ++i) m = fmaxf(m, sc_s[i]);
        float sum = 0.0f;
#pragma unroll
        for (int i = 0; i < Rdim; ++i) sum += __expf(sc_s[i] - m);
        const float e = __expf(sc_s[tid] - m);
        out[b * Rdim + tid] = e / sum;
    }
}

// ---------------------------------------------------------------------------
// Launcher.  Inputs (setup_inputs order):
//   0: seq_embeddings [32,2048,1024] f32
//   1: W              [1,1024,1024]  f32
//   2: b              [1]            f32
//   3: rubric_span    [32,64,2]      i32
//   4: answer_span    [32,2]         i32
//   5: rubric_mask    [32,64]        bool (1 byte)
// out: logits [32,64] f32
// ws : answer_emb (32*1024 f32) | U (32*1024 f32)
// ---------------------------------------------------------------------------
extern "C" void kernel_launch(void* const* d_in, const int* in_sizes, int n_in,
                              void* d_out, int out_size, void* d_ws, size_t ws_size,
                              hipStream_t stream) {
    const float*         seq   = (const float*)d_in[0];
    const float*         W     = (const float*)d_in[1];
    const float*         bias  = (const float*)d_in[2];
    const int*           rspan = (const int*)d_in[3];
    const int*           aspan = (const int*)d_in[4];
    const unsigned char* rmask = (const unsigned char*)d_in[5];
    float*               out   = (float*)d_out;

    float* answer_emb = (float*)d_ws;                    // 32*1024 floats
    float* U          = answer_emb + Bdim * Hdim;        // 32*1024 floats

    // K1: answer pooling
    answer_pool_kernel<<<Bdim, 256, 0, stream>>>(seq, aspan, answer_emb);

    // K2: U = answer_emb @ W^T  via f32 WMMA.  grid: (H/16 tiles, B/16 tiles)
    dim3 g2(Hdim / 16, Bdim / 16);
    bilinear_u_wmma_kernel<<<g2, 32, 0, stream>>>(answer_emb, W, U);

    // K3: scores + masked softmax, one block per batch
    scores_softmax_kernel<<<Bdim, 1024, 0, stream>>>(seq, U, rspan, rmask, bias, out);
}